// CompGCNConv_24489903522003
// MI455X (gfx1250) — compile-verified
//
#include <hip/hip_runtime.h>
#include <hip/hip_bf16.h>

#define HID 256
#define NUM_ENT 100000
#define NUM_REL 50
#define NE 320000      /* edges per direction */
#define TE 640000      /* total edges = 2*NE  */

typedef __attribute__((ext_vector_type(2))) float v2f;
typedef __attribute__((ext_vector_type(8))) float v8f;

// ---------------- utility ----------------
__global__ void zero4_kernel(float4* __restrict__ p, int n4) {
    int i = blockIdx.x * blockDim.x + threadIdx.x;
    if (i < n4) p[i] = make_float4(0.f, 0.f, 0.f, 0.f);
}

// ---------------- degree / norm ----------------
__global__ void deg_kernel(const int* __restrict__ rows, float* __restrict__ deg, int n) {
    int t = blockIdx.x * blockDim.x + threadIdx.x;
    if (t < n) atomicAdd(&deg[rows[t]], 1.0f);
}

__global__ void deginv_kernel(float* __restrict__ deg, int n) {
    int t = blockIdx.x * blockDim.x + threadIdx.x;
    if (t < n) { float d = deg[t]; deg[t] = d > 0.f ? 1.0f / sqrtf(d) : 0.f; }
}

// ---------------- edge scatter: acc[row] += norm * (x[col] * rel[ty]) ----------------
// one wave (32 lanes) per edge, each lane owns 8 consecutive floats of the 256-wide row
__global__ void scatter_kernel(const float* __restrict__ x,
                               const int* __restrict__ rows,
                               const int* __restrict__ cols,
                               const int* __restrict__ types,
                               const float* __restrict__ rel,
                               const float* __restrict__ dinv,
                               float* __restrict__ acc)
{
    int t = blockIdx.x * blockDim.x + threadIdx.x;
    int e = t >> 5, lane = t & 31;
    if (e >= NE) return;
    int r = rows[e], c = cols[e], ty = types[e];
    float nrm = dinv[r] * dinv[c];
    int base = lane * 8;
    const float4* xp = (const float4*)(x   + (size_t)c  * HID + base);
    const float4* rp = (const float4*)(rel + (size_t)ty * HID + base);
    float4 x0 = xp[0], x1 = xp[1];
    float4 r0 = rp[0], r1 = rp[1];
    float* ap = acc + (size_t)r * HID + base;
    atomicAdd(ap + 0, x0.x * r0.x * nrm);
    atomicAdd(ap + 1, x0.y * r0.y * nrm);
    atomicAdd(ap + 2, x0.z * r0.z * nrm);
    atomicAdd(ap + 3, x0.w * r0.w * nrm);
    atomicAdd(ap + 4, x1.x * r1.x * nrm);
    atomicAdd(ap + 5, x1.y * r1.y * nrm);
    atomicAdd(ap + 6, x1.z * r1.z * nrm);
    atomicAdd(ap + 7, x1.w * r1.w * nrm);
}

// ---------------- fp32 WMMA GEMM: D[M x 256] (=|+=) (A [. * kscale]) @ W[256 x 256] ----------------
// block = 128 threads = 4 waves; block handles 16 rows x 256 cols; wave handles 16 x 64.
// A tile layout (16x4 f32): lane m = lane&15; VGPR pair holds K = k0+2*(lane>>4) .. +1 (contiguous float2).
// B tile layout (4x16):     b[0] = W[k][n0+lm], b[1] = W[k+1][n0+lm], same K split by lane>>4.
template <bool SCALE, bool ACCUM>
__global__ void gemm_wmma_kernel(const float* __restrict__ A,
                                 const float* __restrict__ W,
                                 const float* __restrict__ kscale,
                                 float* __restrict__ D)
{
    const int m0   = blockIdx.x * 16;
    const int wave = threadIdx.x >> 5;
    const int lane = threadIdx.x & 31;
    const int n0   = wave * 64;
    const int lm   = lane & 15;
    const int lk   = lane >> 4;

    v8f c0 = {}, c1 = {}, c2 = {}, c3 = {};
    const float* arow = A + (size_t)(m0 + lm) * HID;

    for (int k0 = 0; k0 < HID; k0 += 4) {
        const int k = k0 + 2 * lk;
        v2f a = *(const v2f*)(arow + k);
        if (SCALE) { a[0] *= kscale[k]; a[1] *= kscale[k + 1]; }
        const float* w0 = W + (size_t)k * HID + n0 + lm;
        v2f b0, b1, b2, b3;
        b0[0] = w0[0];  b0[1] = w0[HID + 0];
        b1[0] = w0[16]; b1[1] = w0[HID + 16];
        b2[0] = w0[32]; b2[1] = w0[HID + 32];
        b3[0] = w0[48]; b3[1] = w0[HID + 48];
        c0 = __builtin_amdgcn_wmma_f32_16x16x4_f32(false, a, false, b0, (short)0, c0, false, false);
        c1 = __builtin_amdgcn_wmma_f32_16x16x4_f32(false, a, false, b1, (short)0, c1, false, false);
        c2 = __builtin_amdgcn_wmma_f32_16x16x4_f32(false, a, false, b2, (short)0, c2, false, false);
        c3 = __builtin_amdgcn_wmma_f32_16x16x4_f32(false, a, false, b3, (short)0, c3, false, false);
    }

    // C/D layout: VGPR r -> row m0 + r + 8*lk, col n0 + lm (+16 per tile)
#pragma unroll
    for (int r = 0; r < 8; ++r) {
        const int row = m0 + r + 8 * lk;
        float* dp = D + (size_t)row * HID + n0 + lm;
        if (ACCUM) {
            dp[0]  += c0[r]; dp[16] += c1[r]; dp[32] += c2[r]; dp[48] += c3[r];
        } else {
            dp[0]  = c0[r]; dp[16] = c1[r]; dp[32] = c2[r]; dp[48] = c3[r];
        }
    }
}

// ---------------- batch norm ----------------
__global__ void bn_reduce_kernel(const float* __restrict__ out,
                                 float* __restrict__ sum, float* __restrict__ sumsq)
{
    const int t  = threadIdx.x;           // column 0..255
    const int r0 = blockIdx.x * 256;
    const int rend = min(r0 + 256, NUM_ENT);
    float s = 0.f, s2 = 0.f;
    for (int r = r0; r < rend; ++r) {
        float v = out[(size_t)r * HID + t] * (1.0f / 3.0f);
        s += v; s2 += v * v;
    }
    atomicAdd(&sum[t], s);
    atomicAdd(&sumsq[t], s2);
}

__global__ void bn_finalize_kernel(float* __restrict__ sum, float* __restrict__ sumsq)
{
    int t = threadIdx.x;
    float mean = sum[t] * (1.0f / (float)NUM_ENT);
    float var  = sumsq[t] * (1.0f / (float)NUM_ENT) - mean * mean;
    sum[t]   = mean;
    sumsq[t] = 1.0f / sqrtf(var + 1e-5f);
}

__global__ void bn_apply_kernel(float* __restrict__ out,
                                const float* __restrict__ mean,
                                const float* __restrict__ inv,
                                const float* __restrict__ w,
                                const float* __restrict__ b)
{
    int i = blockIdx.x * blockDim.x + threadIdx.x;   // float4 index
    if (i >= NUM_ENT * (HID / 4)) return;
    int h = (i & 63) << 2;
    float4 v = ((float4*)out)[i];
    v.x = (v.x * (1.0f / 3.0f) - mean[h + 0]) * inv[h + 0] * w[h + 0] + b[h + 0];
    v.y = (v.y * (1.0f / 3.0f) - mean[h + 1]) * inv[h + 1] * w[h + 1] + b[h + 1];
    v.z = (v.z * (1.0f / 3.0f) - mean[h + 2]) * inv[h + 2] * w[h + 2] + b[h + 2];
    v.w = (v.w * (1.0f / 3.0f) - mean[h + 3]) * inv[h + 3] * w[h + 3] + b[h + 3];
    ((float4*)out)[i] = v;
}

// ---------------- rel_out = rel_embed @ w_rel (50x256x256, tiny) ----------------
__global__ void relout_kernel(const float* __restrict__ rel,
                              const float* __restrict__ wr,
                              float* __restrict__ dst)
{
    int n = threadIdx.x, r = blockIdx.x;
    float s = 0.f;
    for (int k = 0; k < HID; ++k)
        s = fmaf(rel[(size_t)r * HID + k], wr[(size_t)k * HID + n], s);
    dst[(size_t)r * HID + n] = s;
}

extern "C" void kernel_launch(void* const* d_in, const int* in_sizes, int n_in,
                              void* d_out, int out_size, void* d_ws, size_t ws_size,
                              hipStream_t stream)
{
    const float* x         = (const float*)d_in[0];
    const int*   ei        = (const int*)d_in[1];
    const int*   et        = (const int*)d_in[2];
    const float* rel_embed = (const float*)d_in[3];
    const float* w_loop    = (const float*)d_in[4];
    const float* w_in      = (const float*)d_in[5];
    const float* w_out     = (const float*)d_in[6];
    const float* w_rel     = (const float*)d_in[7];
    const float* loop_rel  = (const float*)d_in[8];
    const float* bn_w      = (const float*)d_in[9];
    const float* bn_b      = (const float*)d_in[10];

    float* out = (float*)d_out;                 // [100000 x 256] then [50 x 256]
    float* acc     = (float*)d_ws;              // 25,600,000 f
    float* deg_in  = acc + (size_t)NUM_ENT * HID;
    float* deg_out = deg_in + NUM_ENT;
    float* bsum    = deg_out + NUM_ENT;         // 256
    float* bsumsq  = bsum + 256;                // 256

    const int WS_FLOATS = NUM_ENT * HID + 2 * NUM_ENT + 512;

    // 0) zero accumulator + degrees + BN partials
    {
        int n4 = WS_FLOATS / 4;
        zero4_kernel<<<(n4 + 255) / 256, 256, 0, stream>>>((float4*)d_ws, n4);
    }

    // 1) degrees and deg^-1/2 (in-direction rows = ei[0..NE), out-direction rows = ei[NE..2NE))
    deg_kernel<<<(NE + 255) / 256, 256, 0, stream>>>(ei,      deg_in,  NE);
    deg_kernel<<<(NE + 255) / 256, 256, 0, stream>>>(ei + NE, deg_out, NE);
    deginv_kernel<<<(NUM_ENT + 255) / 256, 256, 0, stream>>>(deg_in,  NUM_ENT);
    deginv_kernel<<<(NUM_ENT + 255) / 256, 256, 0, stream>>>(deg_out, NUM_ENT);

    const int scat_blocks = (NE * 32) / 256;    // 40000
    const int gemm_blocks = NUM_ENT / 16;       // 6250, exact

    // 2) in-direction: scatter pre-sums, then out = acc @ w_in
    scatter_kernel<<<scat_blocks, 256, 0, stream>>>(x, ei, ei + TE, et, rel_embed, deg_in, acc);
    gemm_wmma_kernel<false, false><<<gemm_blocks, 128, 0, stream>>>(acc, w_in, nullptr, out);

    // 3) out-direction: re-zero acc, scatter, out += acc @ w_out
    {
        int n4 = (NUM_ENT * HID) / 4;
        zero4_kernel<<<(n4 + 255) / 256, 256, 0, stream>>>((float4*)acc, n4);
    }
    scatter_kernel<<<scat_blocks, 256, 0, stream>>>(x, ei + NE, ei + TE + NE, et + NE,
                                                    rel_embed, deg_out, acc);
    gemm_wmma_kernel<false, true><<<gemm_blocks, 128, 0, stream>>>(acc, w_out, nullptr, out);

    // 4) self-loop: out += (x * loop_rel) @ w_loop
    gemm_wmma_kernel<true, true><<<gemm_blocks, 128, 0, stream>>>(x, w_loop, loop_rel, out);

    // 5) batch norm over entities (1/3 scaling folded into stats + apply)
    bn_reduce_kernel<<<(NUM_ENT + 255) / 256, 256, 0, stream>>>(out, bsum, bsumsq);
    bn_finalize_kernel<<<1, 256, 0, stream>>>(bsum, bsumsq);
    {
        int n4 = NUM_ENT * (HID / 4);
        bn_apply_kernel<<<(n4 + 255) / 256, 256, 0, stream>>>(out, bsum, bsumsq, bn_w, bn_b);
    }

    // 6) rel_out = rel_embed @ w_rel  -> second output region
    relout_kernel<<<NUM_REL, 256, 0, stream>>>(rel_embed, w_rel, out + (size_t)NUM_ENT * HID);
}